// Text2SemanticDecoder_54966991454618
// MI455X (gfx1250) — compile-verified
//
#include <hip/hip_runtime.h>

// Model constants (match reference)
#define LNUM 24
#define BSZ  4
#define NH   16
#define DIM  512
#define SEQ  4096
#define FFD  2048
#define HDD  32           // DIM / NH
#define QKV3 (3 * DIM)    // 1536
#define EPSV 1e-5f
#define SCL  0.1767766952966369f   // 1/sqrt(32)

typedef float v2f __attribute__((ext_vector_type(2)));
typedef float v8f __attribute__((ext_vector_type(8)));

// ---------------------------------------------------------------------------
// GEMV via WMMA f32 16x16x4: out[4,N] = A[4,K] @ W[N,K]^T + bias, opt. ReLU.
// Grid: N/16 blocks of 256 threads (8 waves). Wave w handles K-slice w*K/8.
// Each lane does one b128 load per 2 WMMAs: lanes 0-15 load W[row, k..k+3],
// lanes 16-31 load W[row, k+4..k+7]. WMMA #1 consumes vector elems {0,1}
// (global ks {k,k+1,k+4,k+5}), WMMA #2 elems {2,3} ({k+2,k+3,k+6,k+7});
// A uses the identical K remap, so the K-sum is exact.
// A is zero-padded to M=16 via lane masking; split-K partials reduce in LDS.
// ---------------------------------------------------------------------------
__global__ __launch_bounds__(256)
void gemv16_wmma(const float* __restrict__ A, const float* __restrict__ W,
                 const float* __restrict__ bias, float* __restrict__ out,
                 int K, int N, int relu) {
    const int tid  = threadIdx.x;
    const int wave = __builtin_amdgcn_readfirstlane(tid >> 5);  // uniform
    const int lane = tid & 31;
    const int m    = lane & 15;          // row index within 16 (A) / col (B)
    const int k4   = (lane >> 4) << 2;   // 0 for lanes 0-15, 4 for lanes 16-31
    const int n0   = blockIdx.x * 16;

    const int Kw = K >> 3;               // K per wave
    const int kb = wave * Kw;
    const float amask = (m < BSZ) ? 1.0f : 0.0f;
    const float* arow = A + (size_t)((m < BSZ) ? m : 0) * K;
    const float* wrow = W + (size_t)(n0 + m) * K;

    v8f c = {0.f, 0.f, 0.f, 0.f, 0.f, 0.f, 0.f, 0.f};
    #pragma unroll 2
    for (int k = kb; k < kb + Kw; k += 8) {
        const int kk = k + k4;
        __builtin_prefetch(wrow + kk + 256, 0, 1);   // global_prefetch ~1KB ahead
        float4 bv = *(const float4*)(wrow + kk);
        float4 av = *(const float4*)(arow + kk);
        av.x *= amask; av.y *= amask; av.z *= amask; av.w *= amask;
        v2f a0 = {av.x, av.y}, a1 = {av.z, av.w};
        v2f b0 = {bv.x, bv.y}, b1 = {bv.z, bv.w};
        c = __builtin_amdgcn_wmma_f32_16x16x4_f32(
                false, a0, false, b0, (short)0, c, false, false);
        c = __builtin_amdgcn_wmma_f32_16x16x4_f32(
                false, a1, false, b1, (short)0, c, false, false);
    }

    // Lanes 0-15, VGPR r hold out[M=r][n0+lane]; rows 0..3 are the real batch.
    __shared__ float red[8][4][16];
    if (lane < 16) {
        red[wave][0][lane] = c[0];
        red[wave][1][lane] = c[1];
        red[wave][2][lane] = c[2];
        red[wave][3][lane] = c[3];
    }
    __syncthreads();
    if (tid < 64) {
        const int mm = tid >> 4, nn = tid & 15;
        float s = 0.f;
        #pragma unroll
        for (int w = 0; w < 8; ++w) s += red[w][mm][nn];
        s += bias[n0 + nn];
        if (relu) s = fmaxf(s, 0.f);
        out[(size_t)mm * N + n0 + nn] = s;
    }
}

// ---------------------------------------------------------------------------
// Attention for one (b,h) per block. Cache is NOT modified: position `pos`
// uses the freshly computed k/v from the qkv buffer.
// qkv: [B, 3D];  kc/vc: [B,H,S,HD] (this layer's slice);  out: [B, D]
// ---------------------------------------------------------------------------
__global__ __launch_bounds__(256)
void attn_decode(const float* __restrict__ qkv, const float* __restrict__ kc,
                 const float* __restrict__ vc, float* __restrict__ out,
                 const int* __restrict__ pos_ptr) {
    const int bh = blockIdx.x;
    const int b  = bh / NH, h = bh % NH;
    const int tid = threadIdx.x;
    const int pos = *pos_ptr;
    const int n   = pos + 1;             // attendable positions [0, pos]

    __shared__ float sc[SEQ];            // scores / probabilities
    __shared__ float red[256];
    __shared__ float4 qv4[HDD / 4];
    __shared__ float part[8][HDD];

    const float* qp   = qkv + (size_t)b * QKV3 + h * HDD;
    const float* knew = qkv + (size_t)b * QKV3 + DIM     + h * HDD;
    const float* vnew = qkv + (size_t)b * QKV3 + 2 * DIM + h * HDD;
    const float* kbase = kc + ((size_t)(b * NH + h)) * SEQ * HDD;
    const float* vbase = vc + ((size_t)(b * NH + h)) * SEQ * HDD;

    if (tid < HDD / 4) qv4[tid] = ((const float4*)qp)[tid];
    __syncthreads();

    // ---- scores: one K row (128B = one cache line) per thread -------------
    float lmax = -1e30f;
    for (int j = tid; j < n; j += 256) {
        const float4* kr = (const float4*)((j < pos) ? (kbase + (size_t)j * HDD)
                                                     : knew);
        float d = 0.f;
        #pragma unroll
        for (int t = 0; t < HDD / 4; ++t) {
            float4 kk = kr[t], qq = qv4[t];
            d += kk.x * qq.x + kk.y * qq.y + kk.z * qq.z + kk.w * qq.w;
        }
        d *= SCL;
        sc[j] = d;
        lmax = fmaxf(lmax, d);
    }
    // ---- max reduce -------------------------------------------------------
    red[tid] = lmax; __syncthreads();
    for (int s = 128; s > 0; s >>= 1) {
        if (tid < s) red[tid] = fmaxf(red[tid], red[tid + s]);
        __syncthreads();
    }
    const float mx = red[0]; __syncthreads();
    // ---- exp + sum reduce -------------------------------------------------
    float lsum = 0.f;
    for (int j = tid; j < n; j += 256) {
        float e = __expf(sc[j] - mx);
        sc[j] = e;
        lsum += e;
    }
    red[tid] = lsum; __syncthreads();
    for (int s = 128; s > 0; s >>= 1) {
        if (tid < s) red[tid] += red[tid + s];
        __syncthreads();
    }
    const float inv_sum = 1.0f / red[0];
    // ---- weighted V (wave-coalesced: group g = one wave, lane d) ---------
    const int d = tid & 31, g = tid >> 5;
    float acc = 0.f;
    for (int j = g; j < n; j += 8) {
        const float* vr = (j < pos) ? (vbase + (size_t)j * HDD) : vnew;
        acc += sc[j] * vr[d];
    }
    part[g][d] = acc;
    __syncthreads();
    if (tid < HDD) {
        float s = 0.f;
        #pragma unroll
        for (int w = 0; w < 8; ++w) s += part[w][tid];
        out[(size_t)b * DIM + h * HDD + tid] = s * inv_sum;
    }
}

// ---------------------------------------------------------------------------
// LayerNorm(base + delta) over DIM=512 per batch row. Grid: 4 blocks x 256.
// ---------------------------------------------------------------------------
__global__ __launch_bounds__(256)
void ln_residual(const float* __restrict__ base, const float* __restrict__ delta,
                 const float* __restrict__ w, const float* __restrict__ bvec,
                 float* __restrict__ out) {
    const int b = blockIdx.x, t = threadIdx.x;
    __shared__ float red[256];
    const size_t o = (size_t)b * DIM;
    float v0 = base[o + t]       + delta[o + t];
    float v1 = base[o + t + 256] + delta[o + t + 256];
    red[t] = v0 + v1; __syncthreads();
    for (int s = 128; s > 0; s >>= 1) {
        if (t < s) red[t] += red[t + s];
        __syncthreads();
    }
    const float mean = red[0] * (1.0f / DIM); __syncthreads();
    const float d0 = v0 - mean, d1 = v1 - mean;
    red[t] = d0 * d0 + d1 * d1; __syncthreads();
    for (int s = 128; s > 0; s >>= 1) {
        if (t < s) red[t] += red[t + s];
        __syncthreads();
    }
    const float rstd = rsqrtf(red[0] * (1.0f / DIM) + EPSV);
    out[o + t]       = d0 * rstd * w[t]       + bvec[t];
    out[o + t + 256] = d1 * rstd * w[t + 256] + bvec[t + 256];
}

// ---------------------------------------------------------------------------
extern "C" void kernel_launch(void* const* d_in, const int* in_sizes, int n_in,
                              void* d_out, int out_size, void* d_ws, size_t ws_size,
                              hipStream_t stream) {
    const float* x       = (const float*)d_in[0];
    const float* k_cache = (const float*)d_in[1];
    const float* v_cache = (const float*)d_in[2];
    // d_in[3] attn_mask: unused (decode mask == positions <= kv_cache_len)
    const float* qkv_w = (const float*)d_in[4];
    const float* qkv_b = (const float*)d_in[5];
    const float* out_w = (const float*)d_in[6];
    const float* out_b = (const float*)d_in[7];
    const float* w1    = (const float*)d_in[8];
    const float* b1    = (const float*)d_in[9];
    const float* w2    = (const float*)d_in[10];
    const float* b2    = (const float*)d_in[11];
    const float* nw1   = (const float*)d_in[12];
    const float* nb1   = (const float*)d_in[13];
    const float* nw2   = (const float*)d_in[14];
    const float* nb2   = (const float*)d_in[15];
    const int*   pos   = (const int*)d_in[16];

    float* ws    = (float*)d_ws;
    float* h     = ws;                 // [4,512]
    float* h2    = ws + 2048;          // [4,512]
    float* qkv   = ws + 4096;          // [4,1536]
    float* attno = ws + 10240;         // [4,512]
    float* proj  = ws + 12288;         // [4,512]
    float* mlph  = ws + 14336;         // [4,2048]
    float* mlpo  = ws + 22528;         // [4,512]

    hipMemcpyAsync(h, x, (size_t)BSZ * DIM * sizeof(float),
                   hipMemcpyDeviceToDevice, stream);

    for (int l = 0; l < LNUM; ++l) {
        const size_t kvoff = (size_t)l * BSZ * NH * SEQ * HDD;
        // 1) QKV projection: [4,512] x [1536,512]^T
        gemv16_wmma<<<QKV3 / 16, 256, 0, stream>>>(
            h, qkv_w + (size_t)l * QKV3 * DIM, qkv_b + (size_t)l * QKV3,
            qkv, DIM, QKV3, 0);
        // 2) Attention over cached K/V + fresh k/v at `pos`
        attn_decode<<<BSZ * NH, 256, 0, stream>>>(
            qkv, k_cache + kvoff, v_cache + kvoff, attno, pos);
        // 3) Output projection
        gemv16_wmma<<<DIM / 16, 256, 0, stream>>>(
            attno, out_w + (size_t)l * DIM * DIM, out_b + (size_t)l * DIM,
            proj, DIM, DIM, 0);
        // 4) h2 = LN(h + proj)
        ln_residual<<<BSZ, 256, 0, stream>>>(
            h, proj, nw1 + (size_t)l * DIM, nb1 + (size_t)l * DIM, h2);
        // 5) MLP up + ReLU
        gemv16_wmma<<<FFD / 16, 256, 0, stream>>>(
            h2, w1 + (size_t)l * FFD * DIM, b1 + (size_t)l * FFD,
            mlph, DIM, FFD, 1);
        // 6) MLP down
        gemv16_wmma<<<DIM / 16, 256, 0, stream>>>(
            mlph, w2 + (size_t)l * DIM * FFD, b2 + (size_t)l * DIM,
            mlpo, FFD, DIM, 0);
        // 7) h = LN(h2 + mlpo)
        ln_residual<<<BSZ, 256, 0, stream>>>(
            h2, mlpo, nw2 + (size_t)l * DIM, nb2 + (size_t)l * DIM, h);
    }

    hipMemcpyAsync(d_out, h, (size_t)BSZ * DIM * sizeof(float),
                   hipMemcpyDeviceToDevice, stream);
}